// DeformablePConv_40939628266020
// MI455X (gfx1250) — compile-verified
//
#include <hip/hip_runtime.h>

// ---------------------------------------------------------------------------
// DeformablePConv for MI455X (gfx1250, wave32, WMMA).
// All GEMM stages use V_WMMA_F32_16X16X4_F32 (exact fp32 matrix op on CDNA5).
// Concat intermediate (xc) never touches HBM: it lives in the 320KB WGP LDS.
// Phases 2/3 use multi-tile accumulation per wave so each weight fragment
// feeds 2-8 WMMAs (arithmetic intensity up, global traffic down).
// ---------------------------------------------------------------------------

namespace {
constexpr int Bn    = 8;
constexpr int Cin   = 128;
constexpr int Cout  = 256;
constexpr int Hh    = 128;
constexpr int Wn    = 128;
constexpr int HW    = Hh * Wn;

// Kernel-1 LDS layout (floats): 64-px tile
constexpr int K1_XS_OFF   = 0;                   // [3 rows][128 c][66 w]
constexpr int K1_W_OFF    = 3 * Cin * 66;        // [288 kc][2 khalf][32 n][2 s]
constexpr int K1_B_OFF    = K1_W_OFF + 288 * 2 * 32 * 2;   // [32] bias
constexpr int LDS1_FLOATS = K1_B_OFF + 32;       // 62,240 floats = 248,960 B

// Kernel-2 LDS layout (floats):
constexpr int META_W_OFF = 0;                 // [4][3][64][4] bilinear weights
constexpr int META_I_OFF = 3072;              // [4][3][64][4] bilinear indices (int)
constexpr int ABUF_OFF   = 6144;              // [96 chunks][64 px][4 k]  im2col stage
constexpr int XC_OFF     = 6144 + 24576;      // [64 chunks][64 px][4 k]  concat stage
constexpr int LDS2_FLOATS = 6144 + 24576 + 16384;   // 47,104 floats = 188,416 B
}

typedef __attribute__((ext_vector_type(2))) float v2f;
typedef __attribute__((ext_vector_type(4))) float v4f;
typedef __attribute__((ext_vector_type(8))) float v8f;
typedef __attribute__((ext_vector_type(4))) int   v4i;

__device__ __forceinline__ v8f wmma4(v2f a, v2f b, v8f c) {
  // D = A(16x4) * B(4x16) + C(16x16), full fp32.
  return __builtin_amdgcn_wmma_f32_16x16x4_f32(false, a, false, b, (short)0, c,
                                               false, false);
}

// ---------------------------------------------------------------------------
// Kernel 1: fused 3x3 offset convs (4 branches -> 24 channels) as one WMMA
// GEMM per 64-px tile: M = 64, N = 32 (24 used), K = 9*128 = 1152 (tap-major).
// Weights pre-staged into LDS in WMMA B-fragment layout -> ds_load_b64, no
// divergence, no division in the hot loop.
// ---------------------------------------------------------------------------
extern "C" __global__ __launch_bounds__(256)
void dpc_offset_conv(const float* __restrict__ x,
                     const float* __restrict__ ow0, const float* __restrict__ ob0,
                     const float* __restrict__ ow1, const float* __restrict__ ob1,
                     const float* __restrict__ ow2, const float* __restrict__ ob2,
                     const float* __restrict__ ow3, const float* __restrict__ ob3,
                     float* __restrict__ off)
{
  extern __shared__ float lds[];
  float* xs   = lds + K1_XS_OFF;
  float* wlds = lds + K1_W_OFF;
  float* blds = lds + K1_B_OFF;

  const int tid = threadIdx.x;
  const int b   = blockIdx.x >> 8;               // 256 blocks per batch (H * 2)
  const int rem = blockIdx.x & 255;
  const int h   = rem >> 1;
  const int px0 = (rem & 1) * 64;

  // ---- Bias stage (once, tiny)
  if (tid < 32) {
    float bv = 0.0f;
    if (tid < 24) {
      const float* obp[4] = {ob0, ob1, ob2, ob3};
      bv = obp[tid / 6][tid % 6];
    }
    blds[tid] = bv;
  }

  // ---- Zero-fill W fragment rows n = 24..31
  #pragma unroll 1
  for (int i = 0; i < 36; ++i) {
    int f = i * 256 + tid;                       // 8 * 1152 tasks
    int nn = f / 1152;
    int k  = f - nn * 1152;
    int n  = 24 + nn;
    wlds[(((k >> 2) * 2 + ((k >> 1) & 1)) * 32 + n) * 2 + (k & 1)] = 0.0f;
  }

  // ---- Weight stage: br is loop-uniform -> true global_load, no divergence.
  // Tap-major K: k = tap*128 + c  (must match A-fragment ordering below).
  {
    const float* owp[4] = {ow0, ow1, ow2, ow3};
    #pragma unroll
    for (int br = 0; br < 4; ++br) {
      const float* wp = owp[br];
      #pragma unroll 1
      for (int i = 0; i < 27; ++i) {
        int f   = i * 256 + tid;                 // 6 * 1152 tasks
        int ch  = f / 1152;
        int k   = f - ch * 1152;
        int tap = k >> 7;                        // Cin = 128 -> pure shifts
        int c   = k & 127;
        float v = wp[(ch * Cin + c) * 9 + tap];  // ow[ch][c][ty][tx]
        int n   = br * 6 + ch;
        wlds[(((k >> 2) * 2 + ((k >> 1) & 1)) * 32 + n) * 2 + (k & 1)] = v;
      }
    }
  }

  // ---- x stage: rows h-1..h+1, cols px0-1..px0+64 (66 wide), zero padded
  #pragma unroll 1
  for (int i = 0; i < 99; ++i) {
    int f  = i * 256 + tid;                      // 3*128*66 = 25344 tasks
    int wc = f % 66;
    int g  = f / 66;
    int c  = g & 127;
    int r  = g >> 7;
    int hy = h + r - 1;
    int wx = px0 + wc - 1;
    float v = 0.0f;
    if ((unsigned)hy < (unsigned)Hh && (unsigned)wx < (unsigned)Wn)
      v = x[((b * Cin + c) * Hh + hy) * Wn + wx];
    xs[(r * Cin + c) * 66 + wc] = v;
  }
  __syncthreads();

  // ---- GEMM: 4 m-tiles x 2 n-tiles = 8 tiles, one per wave.
  const int lane  = tid & 31;
  const int wave  = tid >> 5;
  const int nrow  = lane & 15;
  const int khalf = lane >> 4;                   // K pair: 0 -> K{0,1}, 1 -> K{2,3}
  const int mt    = wave >> 1;
  const int nt    = wave & 1;
  const int n     = nt * 16 + nrow;              // 0..31, first 24 valid
  const int px    = mt * 16 + nrow;

  const float bias = blds[n];
  v8f acc;
  #pragma unroll
  for (int v = 0; v < 8; ++v) acc[v] = bias;

  #pragma unroll
  for (int tap = 0; tap < 9; ++tap) {            // ty/tx become constants
    const int ty = tap / 3;
    const int tx = tap - ty * 3;
    // A: a[s] = x[b, c = kc2*4 + khalf*2 + s, h+ty-1, w+tx-1]
    const float* ax = xs + (ty * Cin + khalf * 2) * 66 + px + tx;
    // B: chunk kc = tap*32 + kc2 ; wlds[((kc*2 + khalf)*32 + n)*2 + s]
    const float* wb = wlds + ((tap * 64 + khalf) * 32 + n) * 2;
    #pragma unroll 4
    for (int kc2 = 0; kc2 < 32; ++kc2) {
      v2f a;
      a[0] = ax[kc2 * 264];                      // 264 = 4 channels * 66
      a[1] = ax[kc2 * 264 + 66];
      v2f bm = *(const v2f*)(wb + kc2 * 128);    // ds_load_b64
      acc = wmma4(a, bm, acc);
    }
  }

  if (n < 24) {
    #pragma unroll
    for (int v = 0; v < 8; ++v) {
      int pv = mt * 16 + v + 8 * khalf;          // C layout: lanes16-31 -> M+8
      off[((b * 24 + n) * Hh + h) * Wn + px0 + pv] = acc[v];
    }
  }
}

// ---------------------------------------------------------------------------
// Kernel 2: fused deformable-conv branches + 1x1 conv + BN + SiLU.
// One block = 64 pixels (half a row). All intermediates in LDS.
// ---------------------------------------------------------------------------
extern "C" __global__ __launch_bounds__(256)
void dpc_deform_fused(const float* __restrict__ x,
                      const float* __restrict__ dw0, const float* __restrict__ db0,
                      const float* __restrict__ dw1, const float* __restrict__ db1,
                      const float* __restrict__ dw2, const float* __restrict__ db2,
                      const float* __restrict__ dw3, const float* __restrict__ db3,
                      const float* __restrict__ fw,  const float* __restrict__ fb,
                      const float* __restrict__ gamma, const float* __restrict__ beta,
                      const float* __restrict__ rmean, const float* __restrict__ rvar,
                      const float* __restrict__ off,
                      float* __restrict__ out)
{
  extern __shared__ float lds[];
  float* meta_w = lds + META_W_OFF;
  int*   meta_i = (int*)(lds + META_I_OFF);
  float* Abuf   = lds + ABUF_OFF;
  float* XC     = lds + XC_OFF;
  float* OSTG   = Abuf;                          // reuse A stage for [256][64] out

  const int tid = threadIdx.x;
  const int b   = blockIdx.x >> 8;               // 256 blocks per batch (H * 2)
  const int rem = blockIdx.x & 255;
  const int h   = rem >> 1;
  const int px0 = (rem & 1) * 64;

  const int px_t = tid & 63;                     // task pixel, constant per thread
  const int gq   = tid >> 6;                     // task group 0..3

  // ---- Phase 0: per-(branch, tap, pixel) bilinear corners (channel-invariant)
  #pragma unroll
  for (int i = 0; i < 3; ++i) {
    int g  = i * 4 + gq;                         // 0..11 = br*3 + kk
    int kk = g % 3;
    int br = g / 3;
    const float* ob = off + ((b * 24 + br * 6 + 2 * kk) * Hh + h) * Wn + px0 + px_t;
    float dy = ob[0];
    float dx = ob[HW];
    bool vert = (br & 1);                        // branches 1,3: (3,1) kernels
    float by = vert ? (float)(h + kk - 1) : (float)h;
    float bx = vert ? (float)(px0 + px_t) : (float)(px0 + px_t + kk - 1);
    float py  = by + dy;
    float pxx = bx + dx;
    float fy = floorf(py), fx = floorf(pxx);
    float wy = py - fy, wx = pxx - fx;
    int y0 = (int)fy, x0 = (int)fx;
    int base = (g * 64 + px_t) * 4;
    #pragma unroll
    for (int j = 0; j < 4; ++j) {
      int yj = y0 + (j >> 1);
      int xj = x0 + (j & 1);
      float wjy = (j >> 1) ? wy : (1.0f - wy);
      float wjx = (j & 1) ? wx : (1.0f - wx);
      bool ok = (yj >= 0) && (yj < Hh) && (xj >= 0) && (xj < Wn);
      int yc = min(max(yj, 0), Hh - 1);
      int xc = min(max(xj, 0), Wn - 1);
      meta_i[base + j] = yc * Wn + xc;
      meta_w[base + j] = ok ? (wjy * wjx) : 0.0f;  // zero border
    }
  }
  __syncthreads();

  const int lane  = tid & 31;
  const int wave  = tid >> 5;
  const int nrow  = lane & 15;
  const int khalf = lane >> 4;

  const float* dwp[4] = {dw0, dw1, dw2, dw3};
  const float* dbp[4] = {db0, db1, db2, db3};

  #pragma unroll   // br constant -> direct global pointers, no generic selects
  for (int br = 0; br < 4; ++br) {
    // ---- Phase 1: gather A[k = c*3+kk][px] (x is L2-resident; coalesced in px)
    #pragma unroll 1
    for (int i = 0; i < 96; ++i) {
      int g  = i * 4 + gq;                       // 0..383 = c*3 + kk
      int kk = g % 3;
      int c  = g / 3;
      int mb = ((br * 3 + kk) * 64 + px_t) * 4;
      v4f wv = *(const v4f*)&meta_w[mb];         // ds_load_b128
      v4i iv = *(const v4i*)&meta_i[mb];         // ds_load_b128
      const float* xb = x + (size_t)(b * Cin + c) * HW;
      float v = wv[0] * xb[iv[0]] + wv[1] * xb[iv[1]]
              + wv[2] * xb[iv[2]] + wv[3] * xb[iv[3]];
      int klin = g;                              // = c*3 + kk
      Abuf[((klin >> 2) * 64 + px_t) * 4 + (klin & 3)] = v;
    }
    __syncthreads();

    // ---- Phase 2: branch GEMM (M=64 px, N=64 out, K=384) via WMMA.
    // Wave owns one n-column and TWO m-tiles: each B fragment feeds 2 WMMAs.
    {
      const int nt  = wave & 3;
      const int mb2 = (wave >> 2) * 2;           // m-tile base: 0 or 2
      const int o   = nt * 16 + nrow;
      const float bias = dbp[br][o];
      v8f acc0, acc1;
      #pragma unroll
      for (int v = 0; v < 8; ++v) { acc0[v] = bias; acc1[v] = bias; }
      const float* ap = Abuf + (mb2 * 16 + nrow) * 4 + khalf * 2;
      const float* wp = dwp[br] + o * 384 + khalf * 2;     // global_load_b64
      #pragma unroll 4
      for (int kc = 0; kc < 96; ++kc) {
        v2f bm = *(const v2f*)(wp + kc * 4);
        v2f a0 = *(const v2f*)(ap + kc * 256);             // ds_load_b64
        v2f a1 = *(const v2f*)(ap + kc * 256 + 64);        // m-tile +16 px
        acc0 = wmma4(a0, bm, acc0);
        acc1 = wmma4(a1, bm, acc1);
      }
      // Write into xc stage at concat channel k2 = br*64 + o, chunked layout.
      int k2 = br * 64 + o;
      float* xcw = XC + ((k2 >> 2) * 64) * 4 + (k2 & 3);
      #pragma unroll
      for (int v = 0; v < 8; ++v) {
        int pv = mb2 * 16 + v + 8 * khalf;
        xcw[pv * 4]        = acc0[v];
        xcw[(pv + 16) * 4] = acc1[v];
      }
    }
    __syncthreads();   // protects Abuf (next branch) and XC (phase 3)
  }

  // ---- Phase 3: 1x1 conv GEMM (M=64 px, N=256, K=256) + BN + SiLU.
  // Wave owns TWO n-tiles (wave, wave+8) x all FOUR m-tiles: per K-chunk
  // 2 B loads + 4 A loads feed 8 WMMAs.
  {
    const int o0 = wave * 16 + nrow;             // n-tile q=0
    const int o1 = (wave + 8) * 16 + nrow;       // n-tile q=1
    v8f acc[2][4];
    {
      const float bias0 = fb[o0];
      const float bias1 = fb[o1];
      #pragma unroll
      for (int mt = 0; mt < 4; ++mt)
        #pragma unroll
        for (int v = 0; v < 8; ++v) { acc[0][mt][v] = bias0; acc[1][mt][v] = bias1; }
    }
    const float* ap  = XC + nrow * 4 + khalf * 2;
    const float* wq0 = fw + o0 * 256 + khalf * 2;
    const float* wq1 = fw + o1 * 256 + khalf * 2;
    #pragma unroll 2
    for (int kc = 0; kc < 64; ++kc) {
      v2f b0 = *(const v2f*)(wq0 + kc * 4);      // global_load_b64
      v2f b1 = *(const v2f*)(wq1 + kc * 4);
      #pragma unroll
      for (int mt = 0; mt < 4; ++mt) {
        v2f a = *(const v2f*)(ap + kc * 256 + mt * 64);    // ds_load_b64
        acc[0][mt] = wmma4(a, b0, acc[0][mt]);
        acc[1][mt] = wmma4(a, b1, acc[1][mt]);
      }
    }
    #pragma unroll
    for (int q = 0; q < 2; ++q) {
      const int o = q ? o1 : o0;
      float inv  = gamma[o] * rsqrtf(rvar[o] + 1e-5f);
      float addv = beta[o] - rmean[o] * inv;
      #pragma unroll
      for (int mt = 0; mt < 4; ++mt) {
        #pragma unroll
        for (int v = 0; v < 8; ++v) {
          int pv = mt * 16 + v + 8 * khalf;
          float y = acc[q][mt][v] * inv + addv;
          OSTG[o * 64 + pv] = y / (1.0f + __expf(-y));     // y * sigmoid(y)
        }
      }
    }
  }
  __syncthreads();

  // Coalesced, non-temporal output store (streaming; keep x resident in L2).
  #pragma unroll 1
  for (int i = 0; i < 64; ++i) {
    int o  = i * 4 + gq;                         // 0..255
    float v = OSTG[o * 64 + px_t];
    __builtin_nontemporal_store(
        v, &out[((size_t)(b * Cout + o) * Hh + h) * Wn + px0 + px_t]);
  }
}

// ---------------------------------------------------------------------------
extern "C" void kernel_launch(void* const* d_in, const int* in_sizes, int n_in,
                              void* d_out, int out_size, void* d_ws, size_t ws_size,
                              hipStream_t stream) {
  (void)in_sizes; (void)n_in; (void)out_size; (void)ws_size;
  const float* x     = (const float*)d_in[0];
  const float* ow[4] = {(const float*)d_in[1],  (const float*)d_in[5],
                        (const float*)d_in[9],  (const float*)d_in[13]};
  const float* ob[4] = {(const float*)d_in[2],  (const float*)d_in[6],
                        (const float*)d_in[10], (const float*)d_in[14]};
  const float* dw[4] = {(const float*)d_in[3],  (const float*)d_in[7],
                        (const float*)d_in[11], (const float*)d_in[15]};
  const float* db[4] = {(const float*)d_in[4],  (const float*)d_in[8],
                        (const float*)d_in[12], (const float*)d_in[16]};
  const float* fw    = (const float*)d_in[17];
  const float* fb    = (const float*)d_in[18];
  const float* gamma = (const float*)d_in[19];
  const float* beta  = (const float*)d_in[20];
  const float* rmean = (const float*)d_in[21];
  const float* rvar  = (const float*)d_in[22];
  float* off = (float*)d_ws;                     // 24*B*H*W floats = 12.58 MB

  const size_t lds1 = (size_t)LDS1_FLOATS * sizeof(float);   // 248,960 B
  const size_t lds2 = (size_t)LDS2_FLOATS * sizeof(float);   // 188,416 B
  (void)hipFuncSetAttribute(reinterpret_cast<const void*>(dpc_offset_conv),
                            hipFuncAttributeMaxDynamicSharedMemorySize, (int)lds1);
  (void)hipFuncSetAttribute(reinterpret_cast<const void*>(dpc_deform_fused),
                            hipFuncAttributeMaxDynamicSharedMemorySize, (int)lds2);

  dpc_offset_conv<<<dim3(Bn * Hh * 2), dim3(256), lds1, stream>>>(
      x, ow[0], ob[0], ow[1], ob[1], ow[2], ob[2], ow[3], ob[3], off);

  dpc_deform_fused<<<dim3(Bn * Hh * 2), dim3(256), lds2, stream>>>(
      x, dw[0], db[0], dw[1], db[1], dw[2], db[2], dw[3], db[3],
      fw, fb, gamma, beta, rmean, rvar, off, (float*)d_out);
}